// QuantizedBatchNorm_75393855914350
// MI455X (gfx1250) — compile-verified
//
#include <hip/hip_runtime.h>

// QuantizedBatchNorm (k=8, training fwd) for MI455X / gfx1250.
// x: (N=64, C=256, H=56, W=56) f32.  3 read passes + 1 write pass, ~822 MB
// total traffic -> ~35us at 23.3 TB/s HBM (less: x nearly fits 192MB L2).
// All per-element divisions replaced by uniform/per-channel reciprocals
// precomputed in the tiny finalize kernels, keeping the streaming passes
// VALU-light (dual-issueable) and bandwidth-bound.

#define C_CH 256
#define N_B  64
#define HW   3136        // 56*56
#define S4   784         // HW/4 float4s per (n,c) slice
#define NC   (N_B * C_CH)

#define FLT_BIG 3.402823466e38f

typedef float f4  __attribute__((ext_vector_type(4)));
typedef float v2f __attribute__((ext_vector_type(2)));
typedef float v8f __attribute__((ext_vector_type(8)));
typedef int   v4i __attribute__((ext_vector_type(4)));

typedef __attribute__((address_space(1))) v4i as1_v4i;
typedef __attribute__((address_space(3))) v4i as3_v4i;

// ---------------- CDNA5 async global->LDS staging (ASYNCcnt path) ----------
__device__ __forceinline__ void async_b128(const f4* g, f4* l) {
#if __has_builtin(__builtin_amdgcn_global_load_async_to_lds_b128)
  __builtin_amdgcn_global_load_async_to_lds_b128(
      (as1_v4i*)(unsigned long long)g,              // global addr == AS1 addr
      (as3_v4i*)(unsigned)(unsigned long long)l,    // low 32 bits = LDS offset
      0, 0);
#else
  unsigned loff = (unsigned)(unsigned long long)l;
  asm volatile("global_load_async_to_lds_b128 %0, %1, off"
               :: "v"(loff), "v"(g) : "memory");
#endif
}

__device__ __forceinline__ void wait_async0() {
#if __has_builtin(__builtin_amdgcn_s_wait_asynccnt)
  __builtin_amdgcn_s_wait_asynccnt(0);
#else
  asm volatile("s_wait_asynccnt 0x0" ::: "memory");
#endif
}

__device__ __forceinline__ void stage_slice(const f4* __restrict__ src,
                                            f4* tile, int tid) {
  for (int i = tid; i < S4; i += 256) async_b128(src + i, tile + i);
  wait_async0();
  __syncthreads();
}

// ---------------- wave reductions (wave32) ---------------------------------
// Sum across a wave via v_wmma_f32_16x16x4_f32: A = lane partials (16x4),
// B = ones (4x16)  =>  D[m][n] = p[m] + p[m+16]; finish with in-lane adds
// + one xor-16 shuffle.
__device__ __forceinline__ float waveSum32(float v) {
#if __has_builtin(__builtin_amdgcn_wmma_f32_16x16x4_f32)
  v2f a; a.x = v;    a.y = 0.0f;
  v2f b; b.x = 1.0f; b.y = 1.0f;
  v8f c = {};
  v8f d = __builtin_amdgcn_wmma_f32_16x16x4_f32(
      false, a, false, b, (short)0, c, false, false);
  float s = ((d[0] + d[1]) + (d[2] + d[3])) + ((d[4] + d[5]) + (d[6] + d[7]));
  s += __shfl_xor(s, 16, 32);
  return s;
#else
  for (int o = 16; o; o >>= 1) v += __shfl_xor(v, o, 32);
  return v;
#endif
}

__device__ __forceinline__ float waveMax32(float v) {
  for (int o = 16; o; o >>= 1) v = fmaxf(v, __shfl_xor(v, o, 32));
  return v;
}
__device__ __forceinline__ float waveMin32(float v) {
  for (int o = 16; o; o >>= 1) v = fminf(v, __shfl_xor(v, o, 32));
  return v;
}

// Block-wide (256 thr) max, result broadcast to all threads.
__device__ __forceinline__ float blockMax(float v, float* red) {
  v = waveMax32(v);
  __syncthreads();                       // protect red from prior call
  if ((threadIdx.x & 31) == 0) red[threadIdx.x >> 5] = v;
  __syncthreads();
  float r = red[0];
#pragma unroll
  for (int i = 1; i < 8; ++i) r = fmaxf(r, red[i]);
  return r;
}

// ---------------- fake-quant (matches linear_quant, qmax=255, nzp=128) -----
// Streaming form: reciprocal of scale precomputed (uniform), no per-elem div.
__device__ __forceinline__ float quantR(float v, float scale, float rscale,
                                        float lo, float hi) {
  if (scale == 0.0f) return v;           // passthrough when range == 0
  float cs = fminf(fmaxf(v, lo), hi) - lo;
  return floorf(cs * rscale + 0.5f) * scale + lo;
}
// Finalize form (runs on 256 values only; exact division, matches reference).
__device__ __forceinline__ float quantS(float v, float scale, float lo, float hi) {
  if (scale == 0.0f) return v;
  float cs = fminf(fmaxf(v, lo), hi) - lo;
  return floorf(cs / scale + 0.5f) * scale + lo;
}
__device__ __forceinline__ float quant0(float v, float scale) {
  return quantS(v, scale, -128.0f * scale, 127.0f * scale);
}

// ---------------- Pass A: per-slice sum / min / max of x -------------------
__global__ void __launch_bounds__(256)
pass_a(const float* __restrict__ x, float* __restrict__ sumP,
       float* __restrict__ minP, float* __restrict__ maxP) {
  __shared__ f4 tile[S4];
  __shared__ float rs[8], rmn[8], rmx[8];
  const int tid = threadIdx.x;
  const int s   = blockIdx.x;
  const f4* src = (const f4*)x + (size_t)s * S4;
  stage_slice(src, tile, tid);

  float sm = 0.0f, mn = FLT_BIG, mx = -FLT_BIG;
  for (int i = tid; i < S4; i += 256) {
    f4 v = tile[i];
    sm += (v.x + v.y) + (v.z + v.w);
    mn = fminf(mn, fminf(fminf(v.x, v.y), fminf(v.z, v.w)));
    mx = fmaxf(mx, fmaxf(fmaxf(v.x, v.y), fmaxf(v.z, v.w)));
  }
  sm = waveSum32(sm); mn = waveMin32(mn); mx = waveMax32(mx);
  if ((tid & 31) == 0) { int w = tid >> 5; rs[w] = sm; rmn[w] = mn; rmx[w] = mx; }
  __syncthreads();
  if (tid == 0) {
    float S = 0.0f, Mn = FLT_BIG, Mx = -FLT_BIG;
#pragma unroll
    for (int i = 0; i < 8; ++i) { S += rs[i]; Mn = fminf(Mn, rmn[i]); Mx = fmaxf(Mx, rmx[i]); }
    sumP[s] = S; minP[s] = Mn; maxP[s] = Mx;
  }
}

// ---------------- Finalize A: mean_q, ctr quant params, ctr min/max --------
// par[0..3] = s1, lo1, hi1, 1/s1
__global__ void __launch_bounds__(256)
finalize_a(const float* __restrict__ sumP, const float* __restrict__ minP,
           const float* __restrict__ maxP, const float* __restrict__ run_mean,
           float* __restrict__ mean_q, float* __restrict__ ctrMin,
           float* __restrict__ ctrMax, float* __restrict__ par) {
  __shared__ float red[8];
  const int c = threadIdx.x;                  // one channel per thread
  float S = 0.0f, Mn = FLT_BIG, Mx = -FLT_BIG;
  for (int n = 0; n < N_B; ++n) {
    int i = n * C_CH + c;
    S += sumP[i]; Mn = fminf(Mn, minP[i]); Mx = fmaxf(Mx, maxP[i]);
  }
  const float inv_nhw = 1.0f / (float)(N_B * HW);
  float pm  = 0.875f * run_mean[c] + 0.125f * (S * inv_nhw);
  float mxm = blockMax(fabsf(pm), red);
  float m_q = quant0(pm, 2.0f * mxm / 255.0f);
  mean_q[c] = m_q;

  float cmin = Mn - m_q, cmax = Mx - m_q;     // per-channel bounds of x-mean
  float gmax = blockMax(cmax, red);
  float gmin = -blockMax(-cmin, red);
  float mx1  = fmaxf(fabsf(gmax), fabsf(gmin));
  float s1   = 2.0f * mx1 / 255.0f;
  float lo1  = -128.0f * s1, hi1 = 127.0f * s1;
  ctrMin[c] = quantS(cmin, s1, lo1, hi1);     // q monotone => bounds of ctr
  ctrMax[c] = quantS(cmax, s1, lo1, hi1);
  if (c == 0) { par[0] = s1; par[1] = lo1; par[2] = hi1; par[3] = 1.0f / s1; }
}

// ---------------- Pass B: per-slice sum of ctr^2 ---------------------------
__global__ void __launch_bounds__(256)
pass_b(const float* __restrict__ x, const float* __restrict__ mean_q,
       const float* __restrict__ par, float* __restrict__ sqP) {
  __shared__ f4 tile[S4];
  __shared__ float rs[8];
  const int tid = threadIdx.x;
  const int s = blockIdx.x;
  const int c = s & (C_CH - 1);
  const float m  = mean_q[c];
  const float s1 = par[0], lo1 = par[1], hi1 = par[2], rs1 = par[3];
  const f4* src = (const f4*)x + (size_t)s * S4;
  stage_slice(src, tile, tid);

  float acc = 0.0f;
  for (int i = tid; i < S4; i += 256) {
    f4 v = tile[i];
    float c0 = quantR(v.x - m, s1, rs1, lo1, hi1);
    float c1 = quantR(v.y - m, s1, rs1, lo1, hi1);
    float c2 = quantR(v.z - m, s1, rs1, lo1, hi1);
    float c3 = quantR(v.w - m, s1, rs1, lo1, hi1);
    acc += (c0 * c0 + c1 * c1) + (c2 * c2 + c3 * c3);
  }
  acc = waveSum32(acc);
  if ((tid & 31) == 0) rs[tid >> 5] = acc;
  __syncthreads();
  if (tid == 0) {
    float S = 0.0f;
#pragma unroll
    for (int i = 0; i < 8; ++i) S += rs[i];
    sqP[s] = S;
  }
}

// ---------------- Finalize B: var, inv_den, qw/qb, xn & y quant params -----
// par[4..7] = s2, lo2, hi2, 1/s2 ; par[8..11] = s3, lo3, hi3, 1/s3
__global__ void __launch_bounds__(256)
finalize_b(const float* __restrict__ sqP, const float* __restrict__ run_var,
           const float* __restrict__ weight, const float* __restrict__ bias,
           const float* __restrict__ ctrMin, const float* __restrict__ ctrMax,
           float* __restrict__ inv_den, float* __restrict__ rinv,
           float* __restrict__ qw, float* __restrict__ qb,
           float* __restrict__ par) {
  __shared__ float red[8];
  const int c = threadIdx.x;
  float S = 0.0f;
  for (int n = 0; n < N_B; ++n) S += sqP[n * C_CH + c];
  float pv  = 0.875f * run_var[c] + 0.125f * (S * (1.0f / (float)(N_B * HW)));
  float mxv = blockMax(fabsf(pv), red);
  float var = quant0(pv, 2.0f * mxv / 255.0f);
  float pd  = sqrtf(var + 1e-5f);
  float mxd = blockMax(fabsf(pd), red);
  float idn = quant0(pd, 2.0f * mxd / 255.0f);
  inv_den[c] = idn;
  rinv[c]    = 1.0f / idn;                    // per-channel reciprocal, once

  float xpMax = ctrMax[c] / idn;              // idn >= 0 by construction
  float xpMin = ctrMin[c] / idn;
  float gmax = blockMax(xpMax, red);
  float gmin = -blockMax(-xpMin, red);
  float mx2  = fmaxf(fabsf(gmax), fabsf(gmin));
  float s2   = 2.0f * mx2 / 255.0f, lo2 = -128.0f * s2, hi2 = 127.0f * s2;
  float xnMax = quantS(xpMax, s2, lo2, hi2);
  float xnMin = quantS(xpMin, s2, lo2, hi2);

  float wv  = weight[c];
  float mxw = blockMax(fabsf(wv), red);
  float w_q = quant0(wv, 2.0f * mxw / 255.0f);
  qw[c] = w_q;
  float bv  = bias[c];
  float mxb = blockMax(fabsf(bv), red);       // 0 -> passthrough
  float b_q = quant0(bv, 2.0f * mxb / 255.0f);
  qb[c] = b_q;

  float yMax = (w_q >= 0.0f ? w_q * xnMax : w_q * xnMin) + b_q;
  float yMin = (w_q >= 0.0f ? w_q * xnMin : w_q * xnMax) + b_q;
  float gymax = blockMax(yMax, red);
  float gymin = -blockMax(-yMin, red);
  float mx3 = fmaxf(fabsf(gymax), fabsf(gymin));
  float s3  = 2.0f * mx3 / 255.0f;
  if (c == 0) {
    par[4] = s2; par[5] = lo2; par[6] = hi2; par[7] = 1.0f / s2;
    par[8] = s3; par[9] = -128.0f * s3; par[10] = 127.0f * s3; par[11] = 1.0f / s3;
  }
}

// ---------------- Pass C: y = q3(qw * q2(q1(x-mean)*rinv) + qb) ------------
__global__ void __launch_bounds__(256)
pass_c(const float* __restrict__ x, const float* __restrict__ mean_q,
       const float* __restrict__ rinv, const float* __restrict__ qw,
       const float* __restrict__ qb, const float* __restrict__ par,
       float* __restrict__ out) {
  __shared__ f4 tile[S4];
  const int tid = threadIdx.x;
  const int s = blockIdx.x;
  const int c = s & (C_CH - 1);
  const float m = mean_q[c], rid = rinv[c], w = qw[c], b = qb[c];
  const float s1 = par[0], lo1 = par[1], hi1 = par[2],  rs1 = par[3];
  const float s2 = par[4], lo2 = par[5], hi2 = par[6],  rs2 = par[7];
  const float s3 = par[8], lo3 = par[9], hi3 = par[10], rs3 = par[11];
  const f4* src = (const f4*)x + (size_t)s * S4;
  f4* dst = (f4*)out + (size_t)s * S4;
  stage_slice(src, tile, tid);

  for (int i = tid; i < S4; i += 256) {
    f4 v = tile[i];
    f4 r;
    {
      float ctr = quantR(v.x - m, s1, rs1, lo1, hi1);
      float xn  = quantR(ctr * rid, s2, rs2, lo2, hi2);
      r.x = quantR(w * xn + b, s3, rs3, lo3, hi3);
    }
    {
      float ctr = quantR(v.y - m, s1, rs1, lo1, hi1);
      float xn  = quantR(ctr * rid, s2, rs2, lo2, hi2);
      r.y = quantR(w * xn + b, s3, rs3, lo3, hi3);
    }
    {
      float ctr = quantR(v.z - m, s1, rs1, lo1, hi1);
      float xn  = quantR(ctr * rid, s2, rs2, lo2, hi2);
      r.z = quantR(w * xn + b, s3, rs3, lo3, hi3);
    }
    {
      float ctr = quantR(v.w - m, s1, rs1, lo1, hi1);
      float xn  = quantR(ctr * rid, s2, rs2, lo2, hi2);
      r.w = quantR(w * xn + b, s3, rs3, lo3, hi3);
    }
    __builtin_nontemporal_store(r, dst + i);  // y never re-read: keep L2 for x
  }
}

// ---------------- host entry ----------------------------------------------
extern "C" void kernel_launch(void* const* d_in, const int* in_sizes, int n_in,
                              void* d_out, int out_size, void* d_ws, size_t ws_size,
                              hipStream_t stream) {
  (void)in_sizes; (void)n_in; (void)out_size; (void)ws_size;
  const float* x        = (const float*)d_in[0];
  const float* weight   = (const float*)d_in[1];
  const float* bias     = (const float*)d_in[2];
  const float* run_mean = (const float*)d_in[3];
  const float* run_var  = (const float*)d_in[4];
  float* out = (float*)d_out;

  float* w      = (float*)d_ws;       // ~268 KB scratch, rewritten every call
  float* sumP   = w;
  float* minP   = w + NC;
  float* maxP   = w + 2 * NC;
  float* sqP    = w + 3 * NC;
  float* mean_q = w + 4 * NC;
  float* ctrMin = mean_q + C_CH;
  float* ctrMax = ctrMin + C_CH;
  float* invDen = ctrMax + C_CH;
  float* rinv   = invDen + C_CH;
  float* qwv    = rinv + C_CH;
  float* qbv    = qwv + C_CH;
  float* par    = qbv + C_CH;

  pass_a<<<NC, 256, 0, stream>>>(x, sumP, minP, maxP);
  finalize_a<<<1, 256, 0, stream>>>(sumP, minP, maxP, run_mean,
                                    mean_q, ctrMin, ctrMax, par);
  pass_b<<<NC, 256, 0, stream>>>(x, mean_q, par, sqP);
  finalize_b<<<1, 256, 0, stream>>>(sqP, run_var, weight, bias, ctrMin, ctrMax,
                                    invDen, rinv, qwv, qbv, par);
  pass_c<<<NC, 256, 0, stream>>>(x, mean_q, rinv, qwv, qbv, par, out);
}